// SGPNLoss_35673998361268
// MI455X (gfx1250) — compile-verified
//
#include <hip/hip_runtime.h>
#include <hip/hip_bf16.h>
#include <math.h>

typedef float v2f __attribute__((ext_vector_type(2)));
typedef float v8f __attribute__((ext_vector_type(8)));

#define BATCH 4
#define GY 32
#define GX 64
#define PERB (GY*GX)          // 2048 points per batch
#define NPTS 2048
#define TILES_PER_DIM (NPTS/16)                        // 128 tiles per i/j dim
#define JTILES_PER_WAVE 32                             // j-tiles handled per wave
#define JCHUNKS (TILES_PER_DIM/JTILES_PER_WAVE)        // 4
#define TOTAL_WAVES (BATCH*TILES_PER_DIM*JCHUNKS)      // 2048
#define WAVES_PER_BLOCK 8
#define PAIR_BLOCKS (TOTAL_WAVES/WAVES_PER_BLOCK)      // 256

// accumulator slots (doubles in d_ws):
// 0 exist  1 nonexist  2 xoff  3 yoff  4 n1  5 n0
// 6 sisc   7 disc      8 cnt1  9 cnt2

__global__ void init_acc_kernel(double* acc) {
    if (threadIdx.x < 16) acc[threadIdx.x] = 0.0;
}

// ---------------- lane loss (tiny elementwise reduction) ----------------
__global__ void lane_loss_kernel(const float* __restrict__ conf,
                                 const float* __restrict__ off,
                                 const float* __restrict__ gtp,
                                 double* __restrict__ acc) {
    int idx = blockIdx.x * blockDim.x + threadIdx.x;   // 0..8191
    float exist = 0.f, nonex = 0.f, xo = 0.f, yo = 0.f, c1 = 0.f, c0 = 0.f;
    if (idx < BATCH * PERB) {
        int b = idx / PERB, pos = idx - b * PERB;
        float cg = gtp[b * 3 * PERB + pos];
        float cf = conf[b * PERB + pos];
        float d  = cg - cf;
        float sq = d * d;
        if (cg == 1.0f) {
            exist = sq; c1 = 1.f;
            float dx = gtp[b * 3 * PERB + PERB     + pos] - off[b * 2 * PERB + pos];
            float dy = gtp[b * 3 * PERB + 2 * PERB + pos] - off[b * 2 * PERB + PERB + pos];
            xo = dx * dx; yo = dy * dy;
        } else if (cg == 0.0f) {
            nonex = sq; c0 = 1.f;
        }
    }
    #pragma unroll
    for (int o = 16; o >= 1; o >>= 1) {
        exist += __shfl_xor(exist, o, 32);
        nonex += __shfl_xor(nonex, o, 32);
        xo    += __shfl_xor(xo,    o, 32);
        yo    += __shfl_xor(yo,    o, 32);
        c1    += __shfl_xor(c1,    o, 32);
        c0    += __shfl_xor(c0,    o, 32);
    }
    if ((threadIdx.x & 31) == 0) {
        atomicAdd(&acc[0], (double)exist);
        atomicAdd(&acc[1], (double)nonex);
        atomicAdd(&acc[2], (double)xo);
        atomicAdd(&acc[3], (double)yo);
        atomicAdd(&acc[4], (double)c1);
        atomicAdd(&acc[5], (double)c0);
    }
}

// ---------------- precompute per-point feature powers ----------------
// Q[point][12] = [p1[0..3], p3[0..3], p2[0..3]],  P4[point] = sum_c f^4
__global__ void precompute_kernel(const float* __restrict__ feat,
                                  float* __restrict__ Q,
                                  float* __restrict__ P4) {
    int idx = blockIdx.x * blockDim.x + threadIdx.x;   // point id: b*PERB + n
    if (idx >= BATCH * PERB) return;
    int b = idx / PERB, n = idx - b * PERB;
    const float* fb = feat + (size_t)b * 4 * PERB + n;
    float* q = Q + (size_t)idx * 12;
    float p4 = 0.f;
    #pragma unroll
    for (int c = 0; c < 4; c++) {
        float f  = fb[(size_t)c * PERB];
        float f2 = f * f;
        float f3 = f2 * f;
        q[c]     = f;     // p1
        q[4 + c] = f3;    // p3
        q[8 + c] = f2;    // p2
        p4 += f2 * f2;
    }
    P4[idx] = p4;
}

// ---------------- pairwise instance loss via V_WMMA_F32_16X16X4_F32 ----------------
// s_ij = P4_i + P4_j + sum_k A_i[k]*B_j[k], K=12, done as 3 WMMA k-chunks of 4.
// A = [-4*p3, -4*p1, 6*p2],  B = [p1, p3, p2].
// Each wave owns one 16-row i-stripe and sweeps 32 j-tiles: A-frags + p4_i are
// loaded once and reused; gt_instance (64 MB, streamed once) is read NT.
__global__ void pairwise_kernel(const float* __restrict__ Q,
                                const float* __restrict__ P4,
                                const int* __restrict__ gti,
                                double* __restrict__ acc) {
    __shared__ float s4[4];
    if (threadIdx.x < 4) s4[threadIdx.x] = 0.f;
    __syncthreads();

    const int lane = threadIdx.x & 31;
    const int wave = threadIdx.x >> 5;
    const int W    = blockIdx.x * WAVES_PER_BLOCK + wave;  // 0..2047, exact grid
    const int jc   = W & (JCHUNKS - 1);
    const int row  = W >> 2;                // 0..511 = b*128 + ti
    const int b    = row >> 7;
    const int ti   = row & (TILES_PER_DIM - 1);
    const int i0   = ti * 16;
    const int j0b  = jc * JTILES_PER_WAVE * 16;   // 512 * jc
    const int h    = lane >> 4;             // k-half select per ISA A/B layout
    const int mm   = lane & 15;
    const int k0   = 2 * h;

    // A fragments: fixed for the whole j sweep
    const float* qi = Q + (size_t)(b * NPTS + i0 + mm) * 12;
    v2f a0, a1, a2;
    a0.x = -4.0f * qi[4 + k0]; a0.y = -4.0f * qi[5 + k0];   // -4*p3
    a1.x = -4.0f * qi[k0];     a1.y = -4.0f * qi[k0 + 1];   // -4*p1
    a2.x =  6.0f * qi[8 + k0]; a2.y =  6.0f * qi[9 + k0];   //  6*p2

    float p4i[8];
    #pragma unroll
    for (int v = 0; v < 8; v++)
        p4i[v] = P4[b * NPTS + i0 + v + 8 * h];   // C/D layout: VGPR v -> row v + 8*h

    const size_t gb = (size_t)b * NPTS * NPTS;
    float sisc = 0.f, disc = 0.f, c1 = 0.f, c2 = 0.f;

    for (int jt = 0; jt < JTILES_PER_WAVE; ++jt) {
        const int j0 = j0b + jt * 16;
        const float* qj = Q + (size_t)(b * NPTS + j0 + mm) * 12;
        v2f b0 = {qj[k0],     qj[k0 + 1]};
        v2f b1 = {qj[4 + k0], qj[5 + k0]};
        v2f b2 = {qj[8 + k0], qj[9 + k0]};

        v8f c = {0.f, 0.f, 0.f, 0.f, 0.f, 0.f, 0.f, 0.f};
        c = __builtin_amdgcn_wmma_f32_16x16x4_f32(false, a0, false, b0, (short)0, c, false, false);
        c = __builtin_amdgcn_wmma_f32_16x16x4_f32(false, a1, false, b1, (short)0, c, false, false);
        c = __builtin_amdgcn_wmma_f32_16x16x4_f32(false, a2, false, b2, (short)0, c, false, false);

        const int   jg  = j0 + mm;
        const float p4j = P4[b * NPTS + jg];
        #pragma unroll
        for (int v = 0; v < 8; v++) {
            const int ig = i0 + v + 8 * h;
            float s = c[v] + p4i[v] + p4j;
            float dist = (s > 0.f) ? __builtin_amdgcn_sqrtf(s) : 0.f;
            int g = __builtin_nontemporal_load(gti + gb + (size_t)ig * NPTS + jg);
            float m1 = (g == 1) ? 1.f : 0.f;       // branchless masks -> v_cndmask
            float m2 = (g == 2) ? 1.f : 0.f;
            sisc += m1 * dist;
            c1   += m1;
            float t = 1.0f - dist;
            t = (t > 0.f) ? t : 0.f;
            disc += m2 * t;
            c2   += m2;
        }
    }

    // wave reduce
    #pragma unroll
    for (int o = 16; o >= 1; o >>= 1) {
        sisc += __shfl_xor(sisc, o, 32);
        disc += __shfl_xor(disc, o, 32);
        c1   += __shfl_xor(c1,   o, 32);
        c2   += __shfl_xor(c2,   o, 32);
    }
    // block reduce in LDS (ds_add_f32), one f64 atomic set per block
    if (lane == 0) {
        atomicAdd(&s4[0], sisc);
        atomicAdd(&s4[1], disc);
        atomicAdd(&s4[2], c1);
        atomicAdd(&s4[3], c2);
    }
    __syncthreads();
    if (threadIdx.x == 0) {
        atomicAdd(&acc[6], (double)s4[0]);
        atomicAdd(&acc[7], (double)s4[1]);
        atomicAdd(&acc[8], (double)s4[2]);
        atomicAdd(&acc[9], (double)s4[3]);
    }
}

__global__ void finalize_kernel(const double* __restrict__ acc, float* __restrict__ out) {
    double exist = acc[0] / acc[4];
    double nonex = acc[1] / acc[5];
    double offl  = 0.5 * (acc[2] + acc[3]) / acc[4];
    double inst  = acc[6] / acc[8] + acc[7] / acc[9];
    out[0] = (float)(exist + nonex + offl + inst);
}

extern "C" void kernel_launch(void* const* d_in, const int* in_sizes, int n_in,
                              void* d_out, int out_size, void* d_ws, size_t ws_size,
                              hipStream_t stream) {
    const float* conf = (const float*)d_in[0];
    const float* off  = (const float*)d_in[1];
    const float* feat = (const float*)d_in[2];
    const float* gtp  = (const float*)d_in[3];
    const int*   gti  = (const int*)d_in[4];
    float* out = (float*)d_out;

    double* acc = (double*)d_ws;
    float*  Q   = (float*)((char*)d_ws + 128);
    float*  P4  = (float*)((char*)d_ws + 128 + sizeof(float) * (size_t)BATCH * NPTS * 12);

    init_acc_kernel<<<1, 32, 0, stream>>>(acc);
    lane_loss_kernel<<<(BATCH * PERB + 255) / 256, 256, 0, stream>>>(conf, off, gtp, acc);
    precompute_kernel<<<(BATCH * PERB + 255) / 256, 256, 0, stream>>>(feat, Q, P4);
    // exact grid: 8 waves/block, each wave sweeps 32 j-tiles of one i-stripe
    pairwise_kernel<<<PAIR_BLOCKS, WAVES_PER_BLOCK * 32, 0, stream>>>(Q, P4, gti, acc);
    finalize_kernel<<<1, 1, 0, stream>>>(acc, out);
}